// PFSTLoss_41472204210799
// MI455X (gfx1250) — compile-verified
//
#include <hip/hip_runtime.h>
#include <math.h>

typedef __attribute__((ext_vector_type(2))) float v2f;
typedef __attribute__((ext_vector_type(8))) float v8f;
typedef __attribute__((ext_vector_type(4))) unsigned v4u;
typedef __attribute__((ext_vector_type(8))) unsigned v8u;

#define B_   2
#define C_   19
#define H_   128
#define W_   128
#define CF_  64
#define HC_  32
#define WC_  32
#define NK_  49
#define EPS_ 1e-8f

// ---------------- kernel 0: zero the 3 accumulators ----------------
__global__ void pfst_zero(float* acc) {
  if (threadIdx.x < 3) acc[threadIdx.x] = 0.f;
}

// ------- kernel 1: bilinear x2 upsample (half-pixel, clamped) -------
// One block per (b,c) plane. The 64x64 fp32 source plane is DMA'd into LDS
// by the Tensor Data Mover (tensor_load_to_lds, 2-group D#), then 256
// threads compute the 128x128 outputs from LDS.
__global__ __launch_bounds__(256)
void pfst_upsample(const float* __restrict__ le, float* __restrict__ up) {
  __shared__ float tile[64 * 64];

  const int bc = blockIdx.x;                 // 0 .. B*C-1
  const float* src = le + (size_t)bc * 64 * 64;

  if ((threadIdx.x >> 5) == 0) {             // wave 0 issues the TDM load
    unsigned long long ga = (unsigned long long)(uintptr_t)src;
    unsigned lds_off = (unsigned)(uintptr_t)(void*)&tile[0];  // flat[31:0] == LDS offset

    v4u g0;
    g0.x = 1u;                                         // count=1, no gather
    g0.y = lds_off;                                    // lds_addr
    g0.z = (unsigned)(ga & 0xFFFFFFFFull);             // global_addr[31:0]
    g0.w = (unsigned)((ga >> 32) & 0x01FFFFFFull)      // global_addr[56:32]
         | (2u << 30);                                 // type=2 ("image")

    v8u g1;
    g1[0] = 2u << 16;        // workgroup_mask=0, data_size=2 (4 bytes)
    g1[1] = 64u << 16;       // tensor_dim0[15:0]=64  (bits 63:48)
    g1[2] = 64u << 16;       // tensor_dim0 hi=0, tensor_dim1[15:0]=64 (bits 95:80)
    g1[3] = 64u << 16;       // tensor_dim1 hi=0, tile_dim0=64 (bits 127:112)
    g1[4] = 64u;             // tile_dim1=64, tile_dim2=0
    g1[5] = 64u;             // tensor_dim0_stride[31:0]=64
    g1[6] = 0u;              // stride hi, tensor_dim1_stride lo (unused, 2D tile)
    g1[7] = 0u;

    asm volatile("tensor_load_to_lds %0, %1" :: "s"(g0), "s"(g1) : "memory");
    __builtin_amdgcn_s_wait_tensorcnt(0);
  }
  __syncthreads();

  float* dst = up + (size_t)bc * H_ * W_;
  for (int idx = threadIdx.x; idx < H_ * W_; idx += 256) {
    int w = idx & (W_ - 1);
    int h = idx >> 7;
    float sh = h * 0.5f - 0.25f, sw = w * 0.5f - 0.25f;
    int ih = (int)floorf(sh), iw = (int)floorf(sw);
    float fh = sh - (float)ih, fw = sw - (float)iw;
    int ih0 = ih < 0 ? 0 : (ih > 63 ? 63 : ih);
    int ih1 = ih + 1 < 0 ? 0 : (ih + 1 > 63 ? 63 : ih + 1);
    int iw0 = iw < 0 ? 0 : (iw > 63 ? 63 : iw);
    int iw1 = iw + 1 < 0 ? 0 : (iw + 1 > 63 ? 63 : iw + 1);
    float v00 = tile[ih0 * 64 + iw0], v01 = tile[ih0 * 64 + iw1];
    float v10 = tile[ih1 * 64 + iw0], v11 = tile[ih1 * 64 + iw1];
    dst[idx] = (1.f - fh) * ((1.f - fw) * v00 + fw * v01) +
               fh         * ((1.f - fw) * v10 + fw * v11);
  }
}

// ---------------- kernel 2: main fused kernel ----------------
// 1 wave32 per 32-pixel row segment. WMMA computes the 49x64 dot core.
#define ZB_ (4 * 12 * 64)   // offset of the 64-float zero block in Ft

__global__ __launch_bounds__(32)
void pfst_main(const float* __restrict__ logits_trg,
               const float* __restrict__ x_ema,
               const float* __restrict__ mix_masks,
               const float* __restrict__ logits_up,
               float* __restrict__ acc) {
  __shared__ float Ft[4 * 12 * 64 + 64];  // coarse features [row][col][ch] + zero block
  __shared__ float S[16 * 32];            // WMMA D tile
  __shared__ float invn[48];              // 1/max(||f||,eps) per coarse cell
  __shared__ float simL[32 * 50];         // 49 sims per lane (stride 50)

  const int lane = threadIdx.x;
  const int blk  = blockIdx.x;
  const int tx = blk & 3;
  const int h  = (blk >> 2) & (H_ - 1);
  const int b  = blk >> 9;
  const int wBlock = tx * 32;
  const int c0 = (wBlock - 6) >> 2;   // arithmetic shift == floor div 4
  const int r0 = (h - 6) >> 2;

  // --- stage coarse feature block (4 rows x 12 cols x 64 ch), zero OOB ---
  for (int i = lane; i < 4 * 12 * 64; i += 32) {
    int ch = i & 63;
    int cc = (i >> 6) % 12;
    int rr = i / (12 * 64);
    int gr = r0 + rr, gc = c0 + cc;
    float v = 0.f;
    if (gr >= 0 && gr < HC_ && gc >= 0 && gc < WC_)
      v = x_ema[((size_t)(b * CF_ + ch) * HC_ + gr) * WC_ + gc];
    Ft[(rr * 12 + cc) * 64 + ch] = v;
  }
  for (int i = lane; i < 64; i += 32) Ft[ZB_ + i] = 0.f;  // zero block
  __syncthreads();
  for (int i = lane; i < 48; i += 32) {
    float s = 0.f;
    for (int ch = 0; ch < 64; ++ch) { float v = Ft[i * 64 + ch]; s += v * v; }
    invn[i] = 1.0f / fmaxf(sqrtf(s), EPS_);
  }
  __syncthreads();

  const int myQt = lane >> 4;
  const int myP  = lane & 15;
  const int myPixCol = wBlock + lane;         // pixel owned by this lane
  const int qrow = (h >> 2) - r0;
  const int kofs = (lane >> 4) * 2;           // fp32 WMMA A/B K-offset for hi lanes

  for (int qt = 0; qt < 2; ++qt) {
    const int pixColBase = wBlock + qt * 16;
    // A fragment addressing (row = pixel, K = channel)
    const int am   = lane & 15;
    const int acol = ((pixColBase + am) >> 2) - c0;
    const int aBase = (qrow * 12 + acol) * 64 + kofs;
    // B fragment column (neighbor strip position)
    const int jb  = lane & 15;
    const int fc0 = wBlock - 6 + qt * 16 + jb;
    const int fc1 = fc0 + 16;
    const int cc0 = (fc0 >> 2) - c0;
    const int cc1 = (fc1 >> 2) - c0;

    for (int dyi = 0; dyi < 7; ++dyi) {
      const int fr = h + dyi * 2 - 6;
      const bool rowOK = (fr >= 0 && fr < H_);
      const int nrow = rowOK ? ((fr >> 2) - r0) : 0;
      // invalid taps read the zero block: unconditional, vectorizable LDS loads
      const int bBase0 = ((rowOK && cc0 < 12) ? (nrow * 12 + cc0) * 64 : ZB_) + kofs;
      const int bBase1 = ((rowOK && cc1 < 12) ? (nrow * 12 + cc1) * 64 : ZB_) + kofs;

      v8f a0 = {0.f, 0.f, 0.f, 0.f, 0.f, 0.f, 0.f, 0.f};
      v8f a1 = {0.f, 0.f, 0.f, 0.f, 0.f, 0.f, 0.f, 0.f};
#pragma unroll
      for (int kb = 0; kb < 64; kb += 4) {
        v2f av, b0, b1;
        av.x = Ft[aBase + kb];
        av.y = Ft[aBase + kb + 1];
        b0.x = Ft[bBase0 + kb];
        b0.y = Ft[bBase0 + kb + 1];
        b1.x = Ft[bBase1 + kb];
        b1.y = Ft[bBase1 + kb + 1];
        a0 = __builtin_amdgcn_wmma_f32_16x16x4_f32(false, av, false, b0,
                                                   (short)0, a0, false, false);
        a1 = __builtin_amdgcn_wmma_f32_16x16x4_f32(false, av, false, b1,
                                                   (short)0, a1, false, false);
      }

      // spill D tiles to LDS: D row = v + (lane<16?0:8), col = lane%16
      const int srow = (lane < 16 ? 0 : 8);
#pragma unroll
      for (int v = 0; v < 8; ++v) {
        S[(srow + v) * 32 + (lane & 15)]      = a0[v];
        S[(srow + v) * 32 + 16 + (lane & 15)] = a1[v];
      }
      __syncthreads();

      if (myQt == qt) {
        const int myCc = (myPixCol >> 2) - c0;
        const float invnc = invn[qrow * 12 + myCc];
#pragma unroll
        for (int dxi = 0; dxi < 7; ++dxi) {
          const int j = myP + 2 * dxi;          // band extraction: col = p + 2*dxi
          const float dot = S[myP * 32 + j];
          const int fc = myPixCol + dxi * 2 - 6;
          float inu = 0.f;
          if (rowOK && fc >= 0 && fc < W_)
            inu = invn[nrow * 12 + ((fc >> 2) - c0)];
          simL[lane * 50 + dyi * 7 + dxi] = dot * inu * invnc;
        }
      }
      __syncthreads();
    }
  }

  // ---------------- per-pixel phase (one lane = one pixel) ----------------
  const float* ms = &simL[lane * 50];
  float vmax[9]; int imax[9];
  unsigned long long used = 0ull;
#pragma unroll
  for (int t = 0; t < 9; ++t) {
    float best = -3.0e38f; int bi = 0;
    for (int k = 0; k < NK_; ++k) {
      float s = ms[k];
      if (!((used >> k) & 1ull) && s > best) { best = s; bi = k; }
    }
    vmax[t] = best; imax[t] = bi; used |= (1ull << bi);
  }
  float vmin[8]; int imin[8];
  used = 0ull;
#pragma unroll
  for (int t = 0; t < 8; ++t) {
    float best = 3.0e38f; int bi = 0;
    for (int k = 0; k < NK_; ++k) {
      float s = ms[k];
      if (!((used >> k) & 1ull) && s < best) { best = s; bi = k; }
    }
    vmin[t] = best; imin[t] = bi; used |= (1ull << bi);
  }

  const int w = myPixCol;
  const size_t batchOff = (size_t)b * C_ * H_ * W_;
  float agg[C_];
  float bpos = 0.f, bneg = 0.f;

  // ---- positive branch ----
#pragma unroll
  for (int c = 0; c < C_; ++c) agg[c] = 0.f;
#pragma unroll
  for (int t = 0; t < 9; ++t) {
    const int k = imax[t];
    const int rr = h + (k / 7) * 2 - 6;
    const int cc = w + (k % 7) * 2 - 6;
    const bool ok = (rr >= 0 && rr < H_ && cc >= 0 && cc < W_);
    const int rrc = ok ? rr : 0, ccc = ok ? cc : 0;
    const float* gp = logits_up + batchOff + (size_t)rrc * W_ + ccc;
    const float v = ok ? vmax[t] : 0.f;
#pragma unroll
    for (int c = 0; c < C_; ++c) agg[c] += v * gp[(size_t)c * H_ * W_];
  }
  {
    float mx = -3.0e38f;
#pragma unroll
    for (int c = 0; c < C_; ++c) mx = fmaxf(mx, agg[c]);
    float sum = 0.f;
#pragma unroll
    for (int c = 0; c < C_; ++c) { agg[c] = expf(agg[c] - mx); sum += agg[c]; }
    const float inv = 1.f / sum;
    const float* xt = logits_trg + batchOff + (size_t)h * W_ + w;
#pragma unroll
    for (int c = 0; c < C_; ++c) {
      float x = xt[(size_t)c * H_ * W_];
      float z = agg[c] * inv;
      float ls = fminf(x, 0.f) - log1pf(expf(-fabsf(x)));  // log_sigmoid(x)
      bpos += (1.f - z) * x - ls;
    }
  }

  // ---- negative branch ----
#pragma unroll
  for (int c = 0; c < C_; ++c) agg[c] = 0.f;
#pragma unroll
  for (int t = 0; t < 8; ++t) {
    const int k = imin[t];
    const int rr = h + (k / 7) * 2 - 6;
    const int cc = w + (k % 7) * 2 - 6;
    const bool ok = (rr >= 0 && rr < H_ && cc >= 0 && cc < W_);
    const int rrc = ok ? rr : 0, ccc = ok ? cc : 0;
    const float* gp = logits_up + batchOff + (size_t)rrc * W_ + ccc;
    const float v = ok ? vmin[t] : 0.f;
#pragma unroll
    for (int c = 0; c < C_; ++c) agg[c] += v * gp[(size_t)c * H_ * W_];
  }
  {
    float mx = -3.0e38f;
#pragma unroll
    for (int c = 0; c < C_; ++c) mx = fmaxf(mx, agg[c]);
    float sum = 0.f;
#pragma unroll
    for (int c = 0; c < C_; ++c) { agg[c] = expf(agg[c] - mx); sum += agg[c]; }
    const float inv = 1.f / sum;
    const float* xt = logits_trg + batchOff + (size_t)h * W_ + w;
#pragma unroll
    for (int c = 0; c < C_; ++c) {
      float x = xt[(size_t)c * H_ * W_];
      float z = agg[c] * inv;
      float ls = fminf(x, 0.f) - log1pf(expf(-fabsf(x)));
      bneg += (1.f - z) * x - ls;
    }
  }

  // ---- mask + reduce ----
  const float mix = mix_masks[((size_t)b * H_ + h) * W_ + w];
  const float m = ((1.f - mix) > 0.5f) ? 1.f : 0.f;
  float lp = m * bpos, ln = m * bneg, cnt = m;
#pragma unroll
  for (int off = 16; off > 0; off >>= 1) {
    lp  += __shfl_down(lp,  off, 32);
    ln  += __shfl_down(ln,  off, 32);
    cnt += __shfl_down(cnt, off, 32);
  }
  if (lane == 0) {
    atomicAdd(&acc[0], lp);
    atomicAdd(&acc[1], ln);
    atomicAdd(&acc[2], cnt);
  }
}

// ---------------- kernel 3: finalize ----------------
__global__ void pfst_finalize(const float* __restrict__ acc, float* __restrict__ out) {
  if (threadIdx.x == 0) {
    float denom = acc[2] * (float)C_;
    out[0] = 1.0f * (acc[0] / denom);          // W_POS * loss_pos
    out[1] = 0.1f * (-(acc[1]) / denom);       // W_NEG * loss_neg
  }
}

extern "C" void kernel_launch(void* const* d_in, const int* in_sizes, int n_in,
                              void* d_out, int out_size, void* d_ws, size_t ws_size,
                              hipStream_t stream) {
  (void)in_sizes; (void)n_in; (void)out_size; (void)ws_size;
  const float* logits_trg = (const float*)d_in[0];
  const float* logits_ema = (const float*)d_in[1];
  const float* x_ema      = (const float*)d_in[2];
  const float* mix_masks  = (const float*)d_in[3];
  float* out = (float*)d_out;
  float* ws  = (float*)d_ws;
  float* accv      = ws;        // 3 floats
  float* logits_up = ws + 64;   // B*C*H*W floats (2.49 MB)

  hipLaunchKernelGGL(pfst_zero, dim3(1), dim3(32), 0, stream, accv);
  hipLaunchKernelGGL(pfst_upsample, dim3(B_ * C_), dim3(256), 0, stream,
                     logits_ema, logits_up);
  hipLaunchKernelGGL(pfst_main, dim3(B_ * H_ * 4), dim3(32), 0, stream,
                     logits_trg, x_ema, mix_masks, logits_up, accv);
  hipLaunchKernelGGL(pfst_finalize, dim3(1), dim3(32), 0, stream, accv, out);
}